// KNN_81870666596711
// MI455X (gfx1250) — compile-verified
//
#include <hip/hip_runtime.h>
#include <hip/hip_bf16.h>
#include <float.h>

typedef __attribute__((ext_vector_type(2))) float v2f;
typedef __attribute__((ext_vector_type(8))) float v8f;

// Problem constants (match reference setup_inputs)
constexpr int B_ = 4;
constexpr int N_ = 8192;   // reference points
constexpr int M_ = 8192;   // query points
constexpr int C_ = 3;      // coords
constexpr int K_ = 16;     // top-k

constexpr int QT    = 16;          // queries per wave (one WMMA tile N-dim)
constexpr int WAVES = 8;           // waves per block
constexpr int QPB   = QT * WAVES;  // 128 queries per block

__global__ __launch_bounds__(256)
void knn_wmma_kernel(const float* __restrict__ ref,
                     const float* __restrict__ query,
                     float* __restrict__ out_dist,
                     int*   __restrict__ out_idx)
{
    const int lane = threadIdx.x & 31;
    const int wave = threadIdx.x >> 5;
    const int n    = lane & 15;   // query column within 16x16 tile
    const int hi   = lane >> 4;   // half-wave: 0 -> comps{0,1}/rows 0-7, 1 -> comps{2,rsq}/rows 8-15

    const int tile = blockIdx.x;              // over B_ * (M_/QPB)
    const int b    = tile / (M_ / QPB);
    const int m    = (tile % (M_ / QPB)) * QPB + wave * QT + n;

    // ---- Load this lane's query, build B-matrix column: (-2q0,-2q1,-2q2,1) ----
    const float* qp = query + ((size_t)b * M_ + m) * C_;
    const float q0 = qp[0], q1 = qp[1], q2 = qp[2];
    const float qsq = fmaf(q2, q2, fmaf(q1, q1, q0 * q0));

    v2f bv;
    bv.x = hi ? (-2.0f * q2) : (-2.0f * q0);
    bv.y = hi ? 1.0f         : (-2.0f * q1);

    // ---- Private sorted top-16 (ascending) in VGPRs ----
    float bd[K_];
    int   bi[K_];
#pragma unroll
    for (int t = 0; t < K_; ++t) { bd[t] = FLT_MAX; bi[t] = -1; }

    const float* rbp = ref + (size_t)b * N_ * C_;

    // ---- Stream ref points in chunks of 16; one f32 K=4 WMMA per chunk ----
    for (int rbase = 0; rbase < N_; rbase += 16) {
        const float* rp = rbp + (size_t)(rbase + n) * C_;
        const float r0 = rp[0], r1 = rp[1], r2 = rp[2];
        const float rsq = fmaf(r2, r2, fmaf(r1, r1, r0 * r0));

        // A-matrix row (ref): lanes 0-15 hold K={0,1}, lanes 16-31 hold K={2,3}
        v2f av;
        av.x = hi ? r2  : r0;
        av.y = hi ? rsq : r1;

        v8f c = {};
        // D[M=ref_row, N=query] = rsq - 2*q.r
        c = __builtin_amdgcn_wmma_f32_16x16x4_f32(
                /*neg_a=*/false, av, /*neg_b=*/false, bv,
                /*c_mod=*/(short)0, c, /*reuse_a=*/false, /*reuse_b=*/false);

        // This lane sees ref rows M = v + 8*hi for its query column n.
#pragma unroll
        for (int v = 0; v < 8; ++v) {
            const float d2 = qsq + c[v];
            const int   id = rbase + hi * 8 + v;
            if (d2 < bd[K_ - 1]) {           // reject-fast guard (1 compare)
                float cd = d2; int ci = id;
#pragma unroll
                for (int t = 0; t < K_; ++t) {
                    const bool  s  = cd < bd[t];
                    const float td = bd[t];
                    const int   ti = bi[t];
                    bd[t] = s ? cd : td;  bi[t] = s ? ci : ti;
                    cd    = s ? td : cd;  ci    = s ? ti : ci;
                }
            }
        }
    }

    // ---- Merge half-wave partners: lane L saw rows 0-7, lane L+16 saw rows 8-15 ----
#pragma unroll
    for (int j = 0; j < K_; ++j) {
        const float od = __shfl_xor(bd[j], 16, 32);
        const int   oi = __shfl_xor(bi[j], 16, 32);
        if (od < bd[K_ - 1]) {
            float cd = od; int ci = oi;
#pragma unroll
            for (int t = 0; t < K_; ++t) {
                const bool  s  = cd < bd[t];
                const float td = bd[t];
                const int   ti = bi[t];
                bd[t] = s ? cd : td;  bi[t] = s ? ci : ti;
                cd    = s ? td : cd;  ci    = s ? ti : ci;
            }
        }
    }

    // ---- Lanes 0-15 write the merged result (identical in both halves) ----
    if (hi == 0) {
        const size_t o = ((size_t)b * M_ + m) * K_;
#pragma unroll
        for (int t = 0; t < K_; ++t) {
            out_dist[o + t] = sqrtf(fmaxf(bd[t], 0.0f));
            out_idx[o + t]  = bi[t];
        }
    }
}

extern "C" void kernel_launch(void* const* d_in, const int* in_sizes, int n_in,
                              void* d_out, int out_size, void* d_ws, size_t ws_size,
                              hipStream_t stream)
{
    const float* ref   = (const float*)d_in[0];
    const float* query = (const float*)d_in[1];
    // d_out: [B*M*K] float distances, then [B*M*K] int32 indices (flat concat).
    float* out_dist = (float*)d_out;
    int*   out_idx  = (int*)((float*)d_out + (size_t)B_ * M_ * K_);

    const dim3 grid(B_ * (M_ / QPB));
    const dim3 block(QT * 2 * WAVES);  // 256 threads = 8 wave32s
    knn_wmma_kernel<<<grid, block, 0, stream>>>(ref, query, out_dist, out_idx);
}